// VAR1_28166395527562
// MI455X (gfx1250) — compile-verified
//
#include <hip/hip_runtime.h>

// CDNA5 / gfx1250, wave32. Chunked parallel scan for x_t = A x_{t-1} + d_t.
// DIM_X = 64, DIM_S = 16, T = 262144, chunk C = 256, NCH = 1024.
//
// Structure (recompute scheme, ~96MB total traffic vs 23.3TB/s -> ~4us floor):
//   1) A^256 by 8 repeated squarings (WMMA, log depth)
//   2) phase1: per-chunk zero-start scan -> chunk end vectors e_c only
//   3) sequential combine: s_{c+1} = A^256 s_c + e_c  (1024 short matvecs)
//   4) phase2: re-run chunk scan from s_c, write out = x + mean directly
// Workspace: 2*4096 + 2*NCH*64 floats ~ 0.6 MB.

typedef float v2f __attribute__((ext_vector_type(2)));
typedef float v4f __attribute__((ext_vector_type(4)));
typedef float v8f __attribute__((ext_vector_type(8)));

#define CHUNK 256
#define LOG2C 8
#define USE_ASYNC_LDS 1

// D = A(16x4) x B(4x16) + C(16x16), f32 WMMA.
static __device__ __forceinline__ v8f wmma4(v2f a, v2f b, v8f c) {
  return __builtin_amdgcn_wmma_f32_16x16x4_f32(false, a, false, b, (short)0, c,
                                               false, false);
}

// A-operand fragment: 16x4 tile at (m0,k0) of row-major M[., ld].
static __device__ __forceinline__ v2f loadA(const float* M, int ld, int m0,
                                            int k0, int lm, int hi) {
  const float* p = M + (size_t)(m0 + lm) * ld + k0 + 2 * hi;
  return v2f{p[0], p[1]};
}

static __device__ __forceinline__ void asyncWait() {
#if USE_ASYNC_LDS
  asm volatile("s_wait_asynccnt 0x0" ::: "memory");
#endif
}

// Stage one timestep's U tile (16 chunks x 16 floats) into LDS, coalesced:
// thread f loads floats [sp, sp+1] of chunk cc's input row. 128 thr * 8B = 1KB.
static __device__ __forceinline__ void stageU(const float* __restrict__ inp,
                                              float* ub, int cBase, int C,
                                              int t, int tid) {
  int cc = tid >> 3;
  int sp = (tid & 7) * 2;
  const float* gp = &inp[((size_t)(cBase + cc) * C + t) * 16 + sp];
  float* lp = &ub[cc * 18 + sp];
#if USE_ASYNC_LDS
  asm volatile("global_load_async_to_lds_b64 %0, %1, off"
               :: "v"((unsigned int)(size_t)lp),
                  "v"((unsigned long long)(size_t)gp)
               : "memory");
#else
  v2f u = *(const v2f*)gp;
  lp[0] = u.x;
  lp[1] = u.y;
#endif
}

// ---------------------------------------------------------------------------
// Kernel 0: PA = A
__global__ void init_pow(float* __restrict__ Pw, const float* __restrict__ A) {
  int i = blockIdx.x * blockDim.x + threadIdx.x;
  if (i < 64 * 64) Pw[i] = A[i];
}

// ---------------------------------------------------------------------------
// Kernel 1: dst = src * src (64x64x64 f32 WMMA). One WG, 4 waves.
__global__ void power_sq(float* __restrict__ dst, const float* __restrict__ src) {
  int w = threadIdx.x >> 5, lane = threadIdx.x & 31;
  int lm = lane & 15, hi = lane >> 4;
  int m0 = w * 16;

  v2f af[16];
#pragma unroll
  for (int kk = 0; kk < 16; kk++) af[kk] = loadA(src, 64, m0, 4 * kk, lm, hi);

  for (int nt = 0; nt < 4; nt++) {
    v8f acc[4];
    acc[0] = v8f{}; acc[1] = v8f{}; acc[2] = v8f{}; acc[3] = v8f{};
#pragma unroll
    for (int kk = 0; kk < 16; kk++) {
      int k = 4 * kk + 2 * hi;
      v2f b = {src[(size_t)k * 64 + nt * 16 + lm],
               src[(size_t)(k + 1) * 64 + nt * 16 + lm]};
      acc[kk & 3] = wmma4(af[kk], b, acc[kk & 3]);
    }
    v8f s = (acc[0] + acc[1]) + (acc[2] + acc[3]);
#pragma unroll
    for (int r = 0; r < 8; r++)
      dst[(size_t)(m0 + 8 * hi + r) * 64 + nt * 16 + lm] = s[r];
  }
}

// ---------------------------------------------------------------------------
// Kernel 2/4: chunk scan. One WG = 16 chunks as columns of X[64,16].
// Per step: X <- A*X + Bmat*U_t + intercept, all in WMMA.
// X double-buffered in LDS (one barrier/step); U async-staged one step ahead.
// WRITE_OUT=false: start X=0, emit only chunk ends e_c.
// WRITE_OUT=true : start X=s_c (sArr), emit out rows = X + mean.
template <bool WRITE_OUT>
__global__ void scan_phase(const float* __restrict__ inp,
                           const float* __restrict__ Amat,
                           const float* __restrict__ Bmat,
                           const float* __restrict__ inter,
                           const float* __restrict__ mean,
                           const float* __restrict__ sArr,
                           float* __restrict__ out,
                           float* __restrict__ eArr, int C) {
  __shared__ float Xs[2][64 * 17];  // [buf][m*17 + chunk]
  __shared__ float Ub[2][16 * 18];  // [buf][chunk*18 + s]
  const int tid = threadIdx.x;
  int w = tid >> 5, lane = tid & 31;
  int lm = lane & 15, hi = lane >> 4;
  int m0 = w * 16;
  int cBase = blockIdx.x * 16;

  v2f af[16];
#pragma unroll
  for (int kk = 0; kk < 16; kk++) af[kk] = loadA(Amat, 64, m0, 4 * kk, lm, hi);
  v2f bf[4];
#pragma unroll
  for (int kk = 0; kk < 4; kk++) bf[kk] = loadA(Bmat, 16, m0, 4 * kk, lm, hi);
  v8f icf;
#pragma unroll
  for (int r = 0; r < 8; r++) icf[r] = inter[m0 + 8 * hi + r];
  v8f mf = v8f{};
  if (WRITE_OUT) {
#pragma unroll
    for (int r = 0; r < 8; r++) mf[r] = mean[m0 + 8 * hi + r];
  }

  // init X buffer 0
  if (WRITE_OUT) {
    for (int i = tid; i < 1024; i += 128) {
      int m = i >> 4, cc = i & 15;
      Xs[0][m * 17 + cc] = sArr[(size_t)(cBase + cc) * 64 + m];
    }
  } else {
    for (int i = tid; i < 64 * 17; i += 128) Xs[0][i] = 0.f;
  }
  stageU(inp, Ub[0], cBase, C, 0, tid);
  asyncWait();
  __syncthreads();

  long crow = (long)(cBase + lm) * C;
  int p = 0;
  for (int t = 0; t < C; t++) {
    if (t + 1 < C) stageU(inp, Ub[(t + 1) & 1], cBase, C, t + 1, tid);

    v8f acc[4];
    acc[0] = icf; acc[1] = v8f{}; acc[2] = v8f{}; acc[3] = v8f{};

    // drive: Bmat[64,16] @ U[16,16]
    const float* U = Ub[t & 1];
#pragma unroll
    for (int kk = 0; kk < 4; kk++) {
      int s = 4 * kk + 2 * hi;
      v2f uf = {U[lm * 18 + s], U[lm * 18 + s + 1]};
      acc[kk] = wmma4(bf[kk], uf, acc[kk]);
    }
    // recurrence: A[64,64] @ X[64,16]
    const float* Xc = Xs[p];
#pragma unroll
    for (int kk = 0; kk < 16; kk++) {
      int k = 4 * kk + 2 * hi;
      v2f xb = {Xc[k * 17 + lm], Xc[(k + 1) * 17 + lm]};
      acc[kk & 3] = wmma4(af[kk], xb, acc[kk & 3]);
    }
    v8f xn = (acc[0] + acc[1]) + (acc[2] + acc[3]);

    float* Xn = Xs[p ^ 1];
#pragma unroll
    for (int r = 0; r < 8; r++) Xn[(m0 + 8 * hi + r) * 17 + lm] = xn[r];

    if (WRITE_OUT) {
      v8f o = xn + mf;
      float* op = &out[(crow + t) * 64 + m0 + 8 * hi];
      *(v4f*)op = __builtin_shufflevector(o, o, 0, 1, 2, 3);
      *(v4f*)(op + 4) = __builtin_shufflevector(o, o, 4, 5, 6, 7);
    } else if (t == C - 1) {
      float* ep = &eArr[(size_t)(cBase + lm) * 64 + m0 + 8 * hi];
      *(v4f*)ep = __builtin_shufflevector(xn, xn, 0, 1, 2, 3);
      *(v4f*)(ep + 4) = __builtin_shufflevector(xn, xn, 4, 5, 6, 7);
    }

    asyncWait();      // our staged U for t+1 landed in LDS
    __syncthreads();  // X[p^1] + U visible to all waves
    p ^= 1;
  }
}

// ---------------------------------------------------------------------------
// Kernel 3: sequential chunk combine. s_{c+1} = A^C s_c + e_c; record s_c.
__global__ void combine_seq(const float* __restrict__ PC,
                            const float* __restrict__ eArr,
                            const float* __restrict__ data,
                            const float* __restrict__ mean,
                            float* __restrict__ sArr, int nch) {
  __shared__ float xs[64];
  int m = threadIdx.x;
  float pr[64];
#pragma unroll
  for (int k = 0; k < 64; k++) pr[k] = PC[(size_t)m * 64 + k];
  xs[m] = data[m] - mean[m];
  __syncthreads();
  for (int c = 0; c < nch; c++) {
    sArr[(size_t)c * 64 + m] = xs[m];
    float a0 = 0.f, a1 = 0.f, a2 = 0.f, a3 = 0.f;
#pragma unroll
    for (int k = 0; k < 64; k += 4) {
      a0 = fmaf(pr[k + 0], xs[k + 0], a0);
      a1 = fmaf(pr[k + 1], xs[k + 1], a1);
      a2 = fmaf(pr[k + 2], xs[k + 2], a2);
      a3 = fmaf(pr[k + 3], xs[k + 3], a3);
    }
    float xn = (a0 + a1) + (a2 + a3) + eArr[(size_t)c * 64 + m];
    __syncthreads();
    xs[m] = xn;
    __syncthreads();
  }
}

// ---------------------------------------------------------------------------
extern "C" void kernel_launch(void* const* d_in, const int* in_sizes, int n_in,
                              void* d_out, int out_size, void* d_ws,
                              size_t ws_size, hipStream_t stream) {
  const float* data      = (const float*)d_in[0];  // [1,64]
  const float* inputs    = (const float*)d_in[1];  // [T,16]
  const float* A         = (const float*)d_in[2];  // [64,64]
  const float* B         = (const float*)d_in[3];  // [64,16]
  const float* intercept = (const float*)d_in[4];  // [64]
  const float* mean      = (const float*)d_in[5];  // [1,64]
  int T = in_sizes[1] / 16;
  const int C = CHUNK;
  int nch = T / C;  // 1024

  float* out  = (float*)d_out;
  float* PA   = (float*)d_ws;                  // 4096 floats
  float* PB   = PA + 4096;                     // 4096 floats
  float* eArr = PB + 4096;                     // nch*64
  float* sArr = eArr + (size_t)nch * 64;       // nch*64
  float* bufs[2] = {PA, PB};

  init_pow<<<16, 256, 0, stream>>>(PA, A);
  for (int i = 0; i < LOG2C; i++)  // 8 squarings: A^2 .. A^256, ends in PA
    power_sq<<<1, 128, 0, stream>>>(bufs[(i + 1) & 1], bufs[i & 1]);

  scan_phase<false><<<nch / 16, 128, 0, stream>>>(inputs, A, B, intercept,
                                                  nullptr, nullptr, nullptr,
                                                  eArr, C);
  combine_seq<<<1, 64, 0, stream>>>(PA, eArr, data, mean, sArr, nch);
  scan_phase<true><<<nch / 16, 128, 0, stream>>>(inputs, A, B, intercept, mean,
                                                 sArr, out, nullptr, C);
}